// Gate_36412732735537
// MI455X (gfx1250) — compile-verified
//
#include <hip/hip_runtime.h>

typedef __attribute__((ext_vector_type(2))) float v2f;
typedef __attribute__((ext_vector_type(4))) float v4f;
typedef __attribute__((ext_vector_type(8))) float v8f;

#define BATCH        16
#define C_IN         64
#define KDIM         4096        // 64 * 8 * 8
#define CELLS_PER_B  1024        // 32 * 32
#define TOPK         256

// ---------------------------------------------------------------------------
// Pass 0: Weff[k] = sum_oc W2[oc] * W1[oc, k]   (k = c*64 + ph*8 + pw)
// ---------------------------------------------------------------------------
__global__ void __launch_bounds__(256)
fold_weights(const float* __restrict__ W1, const float* __restrict__ W2,
             float* __restrict__ weff) {
    int k = blockIdx.x * 256 + threadIdx.x;      // 0..4095
    float s = 0.f;
#pragma unroll
    for (int oc = 0; oc < 32; ++oc)
        s += W2[oc] * W1[oc * KDIM + k];
    weff[k] = s;
}

// ---------------------------------------------------------------------------
// Pass 1: gating scores via V_WMMA_F32_16X16X4_F32.
// One wave32 computes 16 consecutive patch cells (M=16), K=4096 in steps of 4.
// B-matrix columns are all Weff (we only need N=0); compute is negligible.
//
// A 16x4 f32 layout (ISA 7.12.2): lanes 0-15 -> M=lane, VGPR0=K+0, VGPR1=K+1;
//                                 lanes 16-31 -> M=lane-16, VGPR0=K+2, VGPR1=K+3.
// B 4x16 f32 layout (row striped across lanes, mirroring C/D half-split):
//                                 VGPR0: K+0 (lanes 0-15) / K+2 (lanes 16-31);
//                                 VGPR1: K+1 / K+3.  Broadcast => per-lane pair
//                                 (Weff[kb+koff], Weff[kb+koff+1]).
// ---------------------------------------------------------------------------
__global__ void __launch_bounds__(256)
patch_gate_wmma(const float* __restrict__ x, const float* __restrict__ weff,
                float* __restrict__ g) {
    __shared__ float sw[KDIM];                   // 16 KB of the 320 KB WGP LDS
    for (int i = threadIdx.x; i < KDIM; i += 256)
        sw[i] = weff[i];
    __syncthreads();

    const int lane  = threadIdx.x & 31;
    const int wave  = threadIdx.x >> 5;
    const int tile  = blockIdx.x * 8 + wave;     // 1024 tiles of 16 cells
    const int cell0 = tile * 16;
    const int m     = lane & 15;
    const int koff  = (lane >> 4) << 1;          // 0 (lanes 0-15) or 2 (16-31)

    const int cell = cell0 + m;
    const int b    = cell >> 10;
    const int rem  = cell & 1023;
    const int h    = rem >> 5;
    const int w    = rem & 31;
    // element addr = b*64*256*256 + c*65536 + (8h+ph)*256 + (8w+pw)
    const float* cbase = x + ((size_t)b << 22) + ((size_t)h << 11) + (w << 3);

    v8f acc = {};
    for (int c = 0; c < C_IN; ++c) {
        const float* ccol = cbase + ((size_t)c << 16);
        const float* wrow = &sw[c << 6];
#pragma unroll
        for (int ph = 0; ph < 8; ++ph) {
#pragma unroll
            for (int pw = 0; pw < 8; pw += 4) {
                const int kk = pw + koff;                         // even => 8B aligned
                v2f a  = *(const v2f*)(ccol + (ph << 8) + kk);    // global_load_b64
                v2f bb = *(const v2f*)(wrow + (ph << 3) + kk);    // ds_load_b64
                acc = __builtin_amdgcn_wmma_f32_16x16x4_f32(
                          false, a, false, bb, (short)0, acc, false, false);
            }
        }
    }

    // D layout: VGPR r, lanes 0-15 -> (M=r, N=lane); lanes 16-31 -> (M=8+r).
    // Column N=0 lives in lane 0 (M=0..7) and lane 16 (M=8..15).
    if ((lane & 15) == 0) {
        const int mbase = (lane >> 4) << 3;
#pragma unroll
        for (int r = 0; r < 8; ++r)
            g[cell0 + mbase + r] = acc[r];
    }
}

// ---------------------------------------------------------------------------
// Pass 2: per-batch top-256 via rank counting (deterministic; tie-break on
// lower index matches lax.top_k). 1024 candidates per batch -> trivial work.
// ---------------------------------------------------------------------------
__global__ void __launch_bounds__(256)
topk_gate(const float* __restrict__ g, float* __restrict__ gate) {
    __shared__ float sv[CELLS_PER_B];
    const int b = blockIdx.x;
    const float* gb = g + b * CELLS_PER_B;
#pragma unroll
    for (int t = 0; t < 4; ++t)
        sv[threadIdx.x + t * 256] = gb[threadIdx.x + t * 256];
    __syncthreads();

    float* ob = gate + b * CELLS_PER_B;
#pragma unroll
    for (int t = 0; t < 4; ++t) {
        const int i = threadIdx.x + t * 256;
        const float vi = sv[i];
        int rank = 0;
        for (int j = 0; j < CELLS_PER_B; ++j) {
            const float vj = sv[j];
            rank += (vj > vi) || (vj == vi && j < i);
        }
        ob[i] = (rank < TOPK) ? vi : 0.f;
    }
}

// ---------------------------------------------------------------------------
// Pass 3: out = in * gate (8x8 upsample + channel broadcast implicit via
// indexing). float4 streaming: ~512 MB of the ~768 MB total traffic.
// ---------------------------------------------------------------------------
__global__ void __launch_bounds__(256)
apply_gate(const float* __restrict__ x, const float* __restrict__ gate,
           float* __restrict__ out) {
    const size_t i4 = (size_t)blockIdx.x * 256 + threadIdx.x;   // 16,777,216 total
    const size_t e  = i4 << 2;                                  // element index
    const int xx = (int)(e & 255);
    const int y  = (int)((e >> 8) & 255);
    const int b  = (int)(e >> 22);
    const float gv = gate[(b << 10) + ((y >> 3) << 5) + (xx >> 3)];
    v4f v = *(const v4f*)(x + e);                               // global_load_b128
    v *= gv;
    *(v4f*)(out + e) = v;                                       // global_store_b128
}

// ---------------------------------------------------------------------------
extern "C" void kernel_launch(void* const* d_in, const int* in_sizes, int n_in,
                              void* d_out, int out_size, void* d_ws, size_t ws_size,
                              hipStream_t stream) {
    const float* x  = (const float*)d_in[0];   // [16,64,256,256]
    const float* W1 = (const float*)d_in[1];   // [32,64,8,8]
    const float* W2 = (const float*)d_in[2];   // [1,32,1,1]
    // d_in[3] = k (=256), folded in as compile-time TOPK
    float* out  = (float*)d_out;

    float* weff = (float*)d_ws;                         // 4096 f32  (16 KB)
    float* g    = weff + KDIM;                          // 16384 f32 (64 KB)
    float* gate = g + BATCH * CELLS_PER_B;              // 16384 f32 (64 KB)

    fold_weights   <<<KDIM / 256,               256, 0, stream>>>(W1, W2, weff);
    patch_gate_wmma<<<128,                      256, 0, stream>>>(x, weff, g);
    topk_gate      <<<BATCH,                    256, 0, stream>>>(g, gate);
    apply_gate     <<<(BATCH * C_IN * 256 * 256) / (4 * 256), 256, 0, stream>>>(x, gate, out);
}